// GGL_33526514713218
// MI455X (gfx1250) — compile-verified
//
#include <hip/hip_runtime.h>
#include <hip/hip_bf16.h>

#define M_CL   1024
#define N_SAMP 64
#define D_DIM  512

typedef __attribute__((ext_vector_type(2))) float v2f;
typedef __attribute__((ext_vector_type(8))) float v8f;

__device__ __forceinline__ float wave_sum32(float v) {
    #pragma unroll
    for (int off = 16; off > 0; off >>= 1) v += __shfl_xor(v, off, 32);
    return v;
}

// ---------------------------------------------------------------------------
// Kernel 1: one block per cluster (256 threads = 8 wave32).
// Computes row inv-norms, renormalized cluster mean (stored to d_ws), and the
// intra-loss partial sum (atomicAdd into acc[0]).
// ---------------------------------------------------------------------------
__global__ __launch_bounds__(256) void GGL_cluster_kernel(
    const float* __restrict__ feat, float* __restrict__ means,
    float* __restrict__ acc) {
    __shared__ float s_inv[N_SAMP];
    __shared__ float s_mean[D_DIM];
    __shared__ float s_red[8];
    __shared__ float s_invmn;

    const int m    = blockIdx.x;
    const int t    = threadIdx.x;
    const int lane = t & 31;
    const int wid  = t >> 5;
    const float* base = feat + (size_t)m * N_SAMP * D_DIM;

    // Phase A: per-row inverse L2 norms (wave w handles rows w, w+8, ...).
    for (int n = wid; n < N_SAMP; n += 8) {
        const float4* rp = (const float4*)(base + n * D_DIM);
        float ss = 0.0f;
        #pragma unroll
        for (int j = 0; j < 4; ++j) {
            float4 v = rp[lane + 32 * j];
            ss += v.x * v.x + v.y * v.y + v.z * v.z + v.w * v.w;
        }
        ss = wave_sum32(ss);
        if (lane == 0) s_inv[n] = 1.0f / fmaxf(sqrtf(ss), 1e-12f);
    }
    __syncthreads();

    // Phase B: mean of normalized rows (each thread owns 2 columns), renorm.
    float mv[2];
    #pragma unroll
    for (int h = 0; h < 2; ++h) {
        const int d = t + h * 256;
        float a = 0.0f;
        for (int n = 0; n < N_SAMP; ++n) a += base[n * D_DIM + d] * s_inv[n];
        mv[h] = a * (1.0f / (float)N_SAMP);
    }
    float ss = mv[0] * mv[0] + mv[1] * mv[1];
    ss = wave_sum32(ss);
    if (lane == 0) s_red[wid] = ss;
    __syncthreads();
    if (t == 0) {
        float tot = 0.0f;
        #pragma unroll
        for (int w = 0; w < 8; ++w) tot += s_red[w];
        s_invmn = 1.0f / fmaxf(sqrtf(tot), 1e-12f);
    }
    __syncthreads();
    const float invmn = s_invmn;
    #pragma unroll
    for (int h = 0; h < 2; ++h) {
        const int d = t + h * 256;
        const float v = mv[h] * invmn;
        s_mean[d] = v;
        means[(size_t)m * D_DIM + d] = v;
    }
    __syncthreads();

    // Phase C: intra partial sum = sum_n (2 - 2 * <row_n / ||row_n||, mean>).
    float rowacc = 0.0f;
    for (int n = wid; n < N_SAMP; n += 8) {
        const float4* rp = (const float4*)(base + n * D_DIM);
        const float4* mp = (const float4*)s_mean;
        float dp = 0.0f;
        #pragma unroll
        for (int j = 0; j < 4; ++j) {
            float4 v = rp[lane + 32 * j];
            float4 w = mp[lane + 32 * j];
            dp += v.x * w.x + v.y * w.y + v.z * w.z + v.w * w.w;
        }
        dp = wave_sum32(dp);
        if (lane == 0) rowacc += 2.0f - 2.0f * (dp * s_inv[n]);
    }
    if (lane == 0) atomicAdd(acc, rowacc);
}

// ---------------------------------------------------------------------------
// Kernel 2: sim = means @ means^T via V_WMMA_F32_16X16X4_F32.
// One wave32 per 16x16 output tile; only tiles with colTile >= rowTile run.
// A layout (32-bit 16x4): lane l -> row (l&15), VGPRs hold K = (l>>4)*2, +1.
// B = means^T has the symmetric layout over the column-tile rows of `means`.
// C/D layout: lane l, VGPR v -> row v + 8*(l>=16), col (l&15).
// ---------------------------------------------------------------------------
__global__ __launch_bounds__(32) void GGL_sim_kernel(
    const float* __restrict__ means, float* __restrict__ acc) {
    const int ct = blockIdx.x;  // column tile
    const int rt = blockIdx.y;  // row tile
    if (ct < rt) return;        // lower-triangular tiles contribute nothing

    const int lane = threadIdx.x;
    const int lm   = lane & 15;
    const int koff = (lane >> 4) * 2;
    const float* arow = means + (size_t)(rt * 16 + lm) * D_DIM + koff;
    const float* brow = means + (size_t)(ct * 16 + lm) * D_DIM + koff;

    v8f c = {};
    for (int k = 0; k < D_DIM; k += 4) {
        v2f a = *(const v2f*)(arow + k);
        v2f b = *(const v2f*)(brow + k);
        c = __builtin_amdgcn_wmma_f32_16x16x4_f32(
            /*neg_a=*/false, a, /*neg_b=*/false, b,
            /*c_mod=*/(short)0, c, /*reuse_a=*/false, /*reuse_b=*/false);
    }

    const int j     = ct * 16 + lm;
    const int ibase = rt * 16 + ((lane >> 4) << 3);
    float s = 0.0f;
    #pragma unroll
    for (int v = 0; v < 8; ++v) {
        const int   i = ibase + v;
        const float x = c[v];
        if (i < j && x > 0.4f) s += x;
    }
    s = wave_sum32(s);
    if (lane == 0) atomicAdd(acc, s);
}

// ---------------------------------------------------------------------------
// Init / finalize
// ---------------------------------------------------------------------------
__global__ void GGL_init_kernel(float* acc) {
    if (threadIdx.x < 2) acc[threadIdx.x] = 0.0f;
}

__global__ void GGL_final_kernel(const float* __restrict__ acc,
                                 float* __restrict__ out) {
    const float intra = acc[0] / ((float)M_CL * (float)N_SAMP);
    const float inter = acc[1] / ((float)M_CL * (float)(M_CL - 1) * 0.5f);
    out[0] = intra + inter;
}

extern "C" void kernel_launch(void* const* d_in, const int* in_sizes, int n_in,
                              void* d_out, int out_size, void* d_ws, size_t ws_size,
                              hipStream_t stream) {
    (void)in_sizes; (void)n_in; (void)out_size; (void)ws_size;
    const float* feat = (const float*)d_in[0];   // [M*N, D] f32
    float* out   = (float*)d_out;                // scalar
    float* acc   = (float*)d_ws;                 // acc[0]=intra sum, acc[1]=inter sum
    float* means = (float*)((char*)d_ws + 256);  // [1024, 512] f32 (2 MB)

    GGL_init_kernel<<<1, 32, 0, stream>>>(acc);
    GGL_cluster_kernel<<<M_CL, 256, 0, stream>>>(feat, means, acc);
    dim3 grid(M_CL / 16, M_CL / 16);
    GGL_sim_kernel<<<grid, 32, 0, stream>>>(means, acc + 1);
    GGL_final_kernel<<<1, 1, 0, stream>>>(acc, out);
}